// LatencyCoding_32521492365347
// MI455X (gfx1250) — compile-verified
//
#include <hip/hip_runtime.h>

// Latency coding for MI455X (gfx1250, wave32).
// Memory-bound: 8 MB in, 256 MB out -> ~11.3 us floor at 23.3 TB/s.
// No matrix contraction exists in this op (WMMA inapplicable). CDNA5 paths:
//  - TDM (tensor_load_to_lds) stages each block's 4KB input tile into LDS,
//    keeping the vector-memory pipe free for the 256MB NT store stream,
//  - NT store cache hints (output > 192MB L2),
//  - wave32 shuffles for the reduction, u32 global atomics, global_prefetch.

typedef float    v4f   __attribute__((ext_vector_type(4)));
typedef unsigned u32x4 __attribute__((ext_vector_type(4)));
typedef int      i32x4 __attribute__((ext_vector_type(4)));
typedef int      i32x8 __attribute__((ext_vector_type(8)));

#define TSTEPS 32
#define TILE_ELEMS 1024  // 256 threads * 4 floats = 4KB per block

// Order-preserving float<->uint map (all finite floats incl. negatives).
__device__ __forceinline__ unsigned mono(float f) {
    unsigned u = __float_as_uint(f);
    return (u & 0x80000000u) ? ~u : (u | 0x80000000u);
}
__device__ __forceinline__ float unmono(unsigned m) {
    return __uint_as_float((m & 0x80000000u) ? (m & 0x7fffffffu) : ~m);
}
__device__ __forceinline__ unsigned umn(unsigned a, unsigned b) { return a < b ? a : b; }
__device__ __forceinline__ unsigned umx(unsigned a, unsigned b) { return a > b ? a : b; }

// --- Kernel 1: reset the min/max accumulators in workspace every call ---
__global__ void lc_init_ws(unsigned* __restrict__ ws) {
    if (threadIdx.x == 0) {
        ws[0] = 0xFFFFFFFFu;  // running min (monotone domain)
        ws[1] = 0x00000000u;  // running max (monotone domain)
    }
}

// --- Kernel 2: global min/max reduction (wave32 shuffles + 1 atomic/block) ---
__global__ __launch_bounds__(256) void lc_minmax(const v4f* __restrict__ in4, int n4,
                                                 unsigned* __restrict__ ws) {
    unsigned mn = 0xFFFFFFFFu, mx = 0u;
    int stride = gridDim.x * blockDim.x;
    for (int i = blockIdx.x * blockDim.x + threadIdx.x; i < n4; i += stride) {
        __builtin_prefetch(&in4[i + stride], 0, 0);  // global_prefetch_b8
        v4f v = in4[i];
#pragma unroll
        for (int c = 0; c < 4; ++c) {
            unsigned m = mono(v[c]);
            mn = umn(mn, m);
            mx = umx(mx, m);
        }
    }
#pragma unroll
    for (int off = 16; off >= 1; off >>= 1) {  // wave32 butterfly
        mn = umn(mn, (unsigned)__shfl_xor((int)mn, off, 32));
        mx = umx(mx, (unsigned)__shfl_xor((int)mx, off, 32));
    }
    __shared__ unsigned smn[8], smx[8];
    int lane = threadIdx.x & 31, wv = threadIdx.x >> 5;
    if (lane == 0) { smn[wv] = mn; smx[wv] = mx; }
    __syncthreads();
    if (threadIdx.x < 8) {
        mn = smn[threadIdx.x];
        mx = smx[threadIdx.x];
#pragma unroll
        for (int off = 4; off >= 1; off >>= 1) {
            mn = umn(mn, (unsigned)__shfl_xor((int)mn, off, 32));
            mx = umx(mx, (unsigned)__shfl_xor((int)mx, off, 32));
        }
        if (threadIdx.x == 0) {
            atomicMin(&ws[0], mn);  // global_atomic_min_u32
            atomicMax(&ws[1], mx);  // global_atomic_max_u32
        }
    }
}

// --- Kernel 3: TDM-staged one-hot expansion ---
// in index  i = b*65536 + p   (p = h*256+w)
// out index   = b*(32*65536) + t*65536 + p
__global__ __launch_bounds__(256) void lc_expand(const float* __restrict__ in,
                                                 float* __restrict__ out,
                                                 const unsigned* __restrict__ ws) {
    __shared__ v4f tile4[TILE_ELEMS / 4];

    unsigned wgBase = blockIdx.x * TILE_ELEMS;  // element offset of this block's tile

#if __has_builtin(__builtin_amdgcn_tensor_load_to_lds) && \
    __has_builtin(__builtin_amdgcn_s_wait_tensorcnt)
    // TDM DMA: global -> LDS, issued once (wave 0 only; branch is wave-uniform
    // under wave32). Frees the vector-memory pipe for the NT store stream.
    if (threadIdx.x < 32) {
        unsigned long long ga = (unsigned long long)(uintptr_t)(in + wgBase);
        unsigned ldsAddr = (unsigned)(uintptr_t)&tile4[0];  // LDS byte offset

        // D# group 0: count=1 | lds_addr | global_addr[56:0] | type=2
        u32x4 g0;
        g0[0] = 1u;                                   // count=1, is_restore=0
        g0[1] = ldsAddr;                              // lds_addr (bytes)
        g0[2] = (unsigned)ga;                         // global_addr[31:0]
        g0[3] = (unsigned)(ga >> 32) | (2u << 30);    // global_addr[56:32] | type=2

        // D# group 1: data_size=2 (4B); tensor_dim0=tile_dim0=1024; 1-D tile.
        i32x8 g1;
        g1[0] = (int)(2u << 16);                      // workgroup_mask=0, data_size=4B
        g1[1] = (int)((TILE_ELEMS & 0xFFFFu) << 16);  // tensor_dim0[15:0] in [31:16]
        g1[2] = (int)((TILE_ELEMS >> 16) | (1u << 16)); // tensor_dim0[31:16] | tensor_dim1=1
        g1[3] = (int)((TILE_ELEMS & 0xFFFFu) << 16);  // tile_dim0=1024 in [31:16]
        g1[4] = 0;                                    // tile_dim1=0, tile_dim2=0 (1-D)
        g1[5] = TILE_ELEMS;                           // tensor_dim0_stride[31:0]
        g1[6] = 0;
        g1[7] = 0;

        i32x4 gz = {0, 0, 0, 0};
#if defined(__clang_major__) && __clang_major__ >= 23
        i32x8 gz8 = {0, 0, 0, 0, 0, 0, 0, 0};
        __builtin_amdgcn_tensor_load_to_lds(g0, g1, gz, gz, gz8, 0);
#else
        __builtin_amdgcn_tensor_load_to_lds(g0, g1, gz, gz, 0);
#endif
        __builtin_amdgcn_s_wait_tensorcnt(0);         // s_wait_tensorcnt 0
    }
    __syncthreads();
    v4f d = tile4[threadIdx.x];                       // ds_load_b128
#else
    // Fallback: direct global load.
    v4f d = ((const v4f*)(in + wgBase))[threadIdx.x];
    (void)tile4;
#endif

    float fmn = unmono(ws[0]);
    float fmx = unmono(ws[1]);
    float denom = fmaxf(fmx - fmn, 1e-12f);
    bool valid = fmx > fmn;

    int st[4];
#pragma unroll
    for (int c = 0; c < 4; ++c) {
        float norm = valid ? (d[c] - fmn) / denom : 0.5f;
        float lat = 1.0f - norm;                      // MAX_LATENCY = 1.0
        int s = (int)(lat * (float)(TSTEPS - 1));     // trunc toward zero
        st[c] = min(max(s, 0), TSTEPS - 1);
    }

    size_t i = (size_t)wgBase + (size_t)threadIdx.x * 4;
    size_t b = i >> 16;          // / 65536
    size_t p = i & 65535u;
    float* obase = out + (b << 21) + p;  // b * 32 * 65536 + p

#pragma unroll
    for (int t = 0; t < TSTEPS; ++t) {
        v4f v;
        v.x = (st[0] == t) ? 1.0f : 0.0f;
        v.y = (st[1] == t) ? 1.0f : 0.0f;
        v.z = (st[2] == t) ? 1.0f : 0.0f;
        v.w = (st[3] == t) ? 1.0f : 0.0f;
        // streaming output (256MB > 192MB L2): global_store_b128 th:TH_STORE_NT
        __builtin_nontemporal_store(v, (v4f*)(obase + ((size_t)t << 16)));
    }
}

extern "C" void kernel_launch(void* const* d_in, const int* in_sizes, int n_in,
                              void* d_out, int out_size, void* d_ws, size_t ws_size,
                              hipStream_t stream) {
    const float* data = (const float*)d_in[0];
    int n = in_sizes[0];       // 32*256*256 = 2,097,152 (multiple of 1024)
    int n4 = n / 4;
    unsigned* ws = (unsigned*)d_ws;

    lc_init_ws<<<1, 32, 0, stream>>>(ws);

    // Reduction: 256 blocks x 256 threads, grid-stride; ~8 float4 per thread.
    lc_minmax<<<256, 256, 0, stream>>>((const v4f*)data, n4, ws);

    // Expansion: one 1024-element tile per block.
    int eblocks = n / TILE_ELEMS;  // 2048
    lc_expand<<<eblocks, 256, 0, stream>>>(data, (float*)d_out, ws);
}